// AlignerOT_63539746177650
// MI455X (gfx1250) — compile-verified
//
#include <hip/hip_runtime.h>
#include <hip/hip_bf16.h>

typedef __attribute__((ext_vector_type(2))) float v2f;
typedef __attribute__((ext_vector_type(8))) float v8f;

#define Hh 512
#define Ss 128
#define Bb 2
#define DSRC 768
#define NITERS 30
#define GAMMA 5000.0f              /* SCALE / REG */
#define ALPHA 2000.0f              /* H * SCALE / S */
#define LOG_MARG (-6.2383246250395077f)  /* -ln(512) */

// ---------------------------------------------------------------------------
// Kernel 1: src[token, h] = sum_d X[token, d] * W[h, d] + bias[h]
// fp32 WMMA 16x16x4. One wave per 16x16 tile. 256x512 output -> 512 waves.
// A layout: lanes 0-15 = M rows, a.x/a.y = K=k0+koff / k0+koff+1, koff=(lane>>4)*2
// B layout (KxN): lanes 0-15 = N cols, b.x/b.y = same K pairing.
// ---------------------------------------------------------------------------
__global__ __launch_bounds__(256) void align_gemm_wmma(
    const float* __restrict__ X, const float* __restrict__ W,
    const float* __restrict__ bias, float* __restrict__ src) {
  const int lane = threadIdx.x & 31;
  const int wave = blockIdx.x * (blockDim.x >> 5) + (threadIdx.x >> 5);
  const int tm = (wave >> 5) << 4;        // 16 M-tiles
  const int tn = (wave & 31) << 4;        // 32 N-tiles
  const int mrow = tm + (lane & 15);
  const int ncol = tn + (lane & 15);
  const int koff = (lane >> 4) << 1;      // 0 or 2
  v8f c = {};
  for (int k0 = 0; k0 < DSRC; k0 += 4) {
    v2f a, bm;
    const float* ap = X + mrow * DSRC + k0 + koff;
    a.x = ap[0]; a.y = ap[1];
    const float* bp = W + ncol * DSRC + k0 + koff;   // B[k][n] = W[n][k]
    bm.x = bp[0]; bm.y = bp[1];
    c = __builtin_amdgcn_wmma_f32_16x16x4_f32(false, a, false, bm,
                                              (short)0, c, false, false);
  }
  const float bb = bias[ncol];
  const int rbase = tm + ((lane >> 4) << 3);
#pragma unroll
  for (int i = 0; i < 8; ++i)
    src[(rbase + i) * Hh + ncol] = c[i] + bb;
}

// ---------------------------------------------------------------------------
// Kernel 2: per-token log-domain Sinkhorn, entirely in LDS.
// logK[j,i] = -GAMMA*(s[i]-t[j])^2 recomputed on the fly (2 VALU ops beats
// 1 MB/token of memory traffic). Two-pass LSE: max pass, then exp-sum pass.
// Finally accumulates ALPHA * plan into psum[b] with coalesced f32 atomics.
// ---------------------------------------------------------------------------
__global__ __launch_bounds__(512) void sinkhorn_kernel(
    const float* __restrict__ src, const float* __restrict__ Y,
    float* __restrict__ psum) {
  __shared__ float s_l[Hh];
  __shared__ float t_l[Hh];
  __shared__ float u_l[Hh];
  __shared__ float v_l[Hh];
  const int tid = threadIdx.x;
  const int tok = blockIdx.x;                 // flat b*S + s
  s_l[tid] = src[tok * Hh + tid];
  t_l[tid] = Y[tok * Hh + tid];
  u_l[tid] = 0.0f;
  v_l[tid] = 0.0f;
  __syncthreads();
  const float tj = t_l[tid];
  const float si = s_l[tid];

  for (int it = 0; it < NITERS; ++it) {
    // u[j] = LOG_MARG - LSE_i( v[i] - GAMMA*(s[i]-t[j])^2 )
    float m = -3.0e38f;
    for (int i = 0; i < Hh; ++i) {
      float d = s_l[i] - tj;
      float x = fmaf(d, -GAMMA * d, v_l[i]);
      m = fmaxf(m, x);
    }
    float acc = 0.0f;
    for (int i = 0; i < Hh; ++i) {
      float d = s_l[i] - tj;
      float x = fmaf(d, -GAMMA * d, v_l[i]);
      acc += __expf(x - m);
    }
    float unew = LOG_MARG - (m + __logf(acc));
    __syncthreads();
    u_l[tid] = unew;
    __syncthreads();

    // v[i] = LOG_MARG - LSE_j( u[j] - GAMMA*(s[i]-t[j])^2 )
    m = -3.0e38f;
    for (int j = 0; j < Hh; ++j) {
      float d = si - t_l[j];
      float x = fmaf(d, -GAMMA * d, u_l[j]);
      m = fmaxf(m, x);
    }
    acc = 0.0f;
    for (int j = 0; j < Hh; ++j) {
      float d = si - t_l[j];
      float x = fmaf(d, -GAMMA * d, u_l[j]);
      acc += __expf(x - m);
    }
    float vnew = LOG_MARG - (m + __logf(acc));
    __syncthreads();
    v_l[tid] = vnew;
    __syncthreads();
  }

  // plan[j,i] = exp(logK[j,i] + u[j] + v[i]); accumulate ALPHA*plan into
  // psum[b]. Loop trip = row j, lane = column i=tid -> coalesced atomics.
  const int b = tok >> 7;
  float* P = psum + b * Hh * Hh;
  const float vi = v_l[tid];
  for (int j = 0; j < Hh; ++j) {
    float d = si - t_l[j];
    float p = __expf(fmaf(d, -GAMMA * d, u_l[j] + vi));
    atomicAdd(&P[j * Hh + tid], ALPHA * p);
  }
}

// ---------------------------------------------------------------------------
// Kernel 3: out[b] = src[b] (128x512)  @  (psum[b] + delta) (512x512)
// psum already carries the H*SCALE/S scaling; delta fused into B-tile loads.
// 256 tiles/batch x 2 batches = 512 waves.
// ---------------------------------------------------------------------------
__global__ __launch_bounds__(256) void out_gemm_wmma(
    const float* __restrict__ src, const float* __restrict__ psum,
    const float* __restrict__ delta, float* __restrict__ out) {
  const int lane = threadIdx.x & 31;
  const int wave = blockIdx.x * (blockDim.x >> 5) + (threadIdx.x >> 5);
  const int b = wave >> 8;
  const int t = wave & 255;               // 8 M-tiles x 32 N-tiles
  const int tm = (t >> 5) << 4;
  const int tn = (t & 31) << 4;
  const int mrow = tm + (lane & 15);
  const int ncol = tn + (lane & 15);
  const int koff = (lane >> 4) << 1;
  const float* A = src + b * Ss * Hh;
  const float* P = psum + b * Hh * Hh;
  v8f c = {};
  for (int k0 = 0; k0 < Hh; k0 += 4) {
    v2f a, bm;
    const float* ap = A + mrow * Hh + k0 + koff;
    a.x = ap[0]; a.y = ap[1];
    const int bi = (k0 + koff) * Hh + ncol;
    bm.x = P[bi] + delta[bi];
    bm.y = P[bi + Hh] + delta[bi + Hh];
    c = __builtin_amdgcn_wmma_f32_16x16x4_f32(false, a, false, bm,
                                              (short)0, c, false, false);
  }
  const int rbase = tm + ((lane >> 4) << 3);
  float* O = out + b * Ss * Hh;
#pragma unroll
  for (int i = 0; i < 8; ++i)
    O[(rbase + i) * Hh + ncol] = c[i];
}

extern "C" void kernel_launch(void* const* d_in, const int* in_sizes, int n_in,
                              void* d_out, int out_size, void* d_ws, size_t ws_size,
                              hipStream_t stream) {
  const float* X     = (const float*)d_in[0];   // (2,128,768)
  const float* Y     = (const float*)d_in[1];   // (2,128,512)
  const float* W     = (const float*)d_in[2];   // (512,768)
  const float* bias  = (const float*)d_in[3];   // (512,)
  const float* delta = (const float*)d_in[4];   // (512,512)
  float* out = (float*)d_out;                   // (2,128,512)

  float* src  = (float*)d_ws;                   // 256*512 f32 = 512 KB
  float* psum = src + Bb * Ss * Hh;             // 2*512*512 f32 = 2 MB

  hipMemsetAsync(psum, 0, (size_t)Bb * Hh * Hh * sizeof(float), stream);
  align_gemm_wmma<<<64, 256, 0, stream>>>(X, W, bias, src);
  sinkhorn_kernel<<<Bb * Ss, 512, 0, stream>>>(src, Y, psum);
  out_gemm_wmma<<<64, 256, 0, stream>>>(src, psum, delta, out);
}